// SAGE_90477781058261
// MI455X (gfx1250) — compile-verified
//
#include <hip/hip_runtime.h>

typedef float v2f __attribute__((ext_vector_type(2)));
typedef float v8f __attribute__((ext_vector_type(8)));

#define N_NODES 100000
#define N_EDGES 1200000
#define D_IN 64
#define D_HID 128
#define D_OUT 64

// ---------------------------------------------------------------------------
// Edge scatter: agg[dst] += x[src] * w   (+ deg[dst] += 1 once per edge)
// One thread handles 4 consecutive features (float4 load, 4 f32 HW atomics).
// ---------------------------------------------------------------------------
__global__ void __launch_bounds__(256) edge_scatter_kernel(
    const float* __restrict__ feat, const int* __restrict__ src,
    const int* __restrict__ dst, const float* __restrict__ ew,
    float* __restrict__ agg, float* __restrict__ deg,
    int D, int addDeg)
{
    const int tpe = D >> 2;  // threads per edge
    long long tid = (long long)blockIdx.x * blockDim.x + threadIdx.x;
    long long e = tid / tpe;
    int f = (int)(tid % tpe) << 2;
    if (e >= N_EDGES) return;

    int s = src[e];
    int d = dst[e];
    float w = ew[e];

    const float4 xv = *(const float4*)(feat + (size_t)s * D + f);
    float* ap = agg + (size_t)d * D + f;
    __hip_atomic_fetch_add(ap + 0, xv.x * w, __ATOMIC_RELAXED, __HIP_MEMORY_SCOPE_AGENT);
    __hip_atomic_fetch_add(ap + 1, xv.y * w, __ATOMIC_RELAXED, __HIP_MEMORY_SCOPE_AGENT);
    __hip_atomic_fetch_add(ap + 2, xv.z * w, __ATOMIC_RELAXED, __HIP_MEMORY_SCOPE_AGENT);
    __hip_atomic_fetch_add(ap + 3, xv.w * w, __ATOMIC_RELAXED, __HIP_MEMORY_SCOPE_AGENT);
    if (addDeg && f == 0)
        __hip_atomic_fetch_add(deg + d, 1.0f, __ATOMIC_RELAXED, __HIP_MEMORY_SCOPE_AGENT);
}

// ---------------------------------------------------------------------------
// Normalize: agg[n][:] /= max(deg[n], 1)   (float4 per thread)
// ---------------------------------------------------------------------------
__global__ void __launch_bounds__(256) normalize_kernel(
    float* __restrict__ agg, const float* __restrict__ deg, int D)
{
    const int vpr = D >> 2;  // float4's per row
    long long i = (long long)blockIdx.x * blockDim.x + threadIdx.x;
    long long total = (long long)N_NODES * vpr;
    if (i >= total) return;
    int n = (int)(i / vpr);
    float inv = 1.0f / fmaxf(deg[n], 1.0f);
    float4* p = (float4*)agg + i;
    float4 v = *p;
    v.x *= inv; v.y *= inv; v.z *= inv; v.w *= inv;
    *p = v;
}

// ---------------------------------------------------------------------------
// Fused SAGE GEMM:  out = act( A1 @ W1 + A2 @ W2 + bias )
// A1,A2: [N_NODES, K] row-major.  W1,W2: [K, NOUT] row-major.
// One wave32 computes one 16x16 output tile via V_WMMA_F32_16X16X4_F32,
// chaining the accumulator across both GEMMs (K/4 + K/4 WMMA ops).
//
// Lane layout (ISA 7.12.2, 32-bit, wave32):
//   A 16x4 : VGPR0 = A[m][kb],  VGPR1 = A[m][kb+1],  m = lane&15, kb = (lane>>4)*2
//   B 4x16 : VGPR0 = B[kb][n],  VGPR1 = B[kb+1][n],  n = lane&15
//   C 16x16: VGPR i = C[i + 8*(lane>>4)][lane&15]
// ---------------------------------------------------------------------------
template<int K, int NOUT, bool RELU>
__global__ void __launch_bounds__(128) sage_gemm_kernel(
    const float* __restrict__ A1, const float* __restrict__ W1,
    const float* __restrict__ A2, const float* __restrict__ W2,
    const float* __restrict__ bias, float* __restrict__ out)
{
    constexpr int NT = NOUT / 16;            // N tiles
    constexpr int MT = N_NODES / 16;         // M tiles (100000/16 = 6250, exact)
    const int wave = threadIdx.x >> 5;
    const int lane = threadIdx.x & 31;
    const int tile = blockIdx.x * 4 + wave;  // 4 waves / block, 1 tile / wave
    if (tile >= MT * NT) return;             // uniform per wave -> EXEC all-1s inside

    const int mt = tile / NT;
    const int nt = tile % NT;
    const int m    = mt * 16 + (lane & 15);
    const int n    = nt * 16 + (lane & 15);
    const int koff = (lane >> 4) << 1;       // 0 or 2

    const float* a1 = A1 + (size_t)m * K + koff;
    const float* a2 = A2 + (size_t)m * K + koff;
    __builtin_prefetch(a2, 0, 1);            // warm second-operand row (global_prefetch_b8)

    v8f c = {};
#pragma unroll
    for (int k0 = 0; k0 < K; k0 += 4) {
        v2f a = *(const v2f*)(a1 + k0);
        v2f b;
        b.x = W1[(k0 + koff) * NOUT + n];
        b.y = W1[(k0 + koff + 1) * NOUT + n];
        c = __builtin_amdgcn_wmma_f32_16x16x4_f32(false, a, false, b,
                                                  (short)0, c, false, false);
    }
#pragma unroll
    for (int k0 = 0; k0 < K; k0 += 4) {
        v2f a = *(const v2f*)(a2 + k0);
        v2f b;
        b.x = W2[(k0 + koff) * NOUT + n];
        b.y = W2[(k0 + koff + 1) * NOUT + n];
        c = __builtin_amdgcn_wmma_f32_16x16x4_f32(false, a, false, b,
                                                  (short)0, c, false, false);
    }

    const float bv = bias[n];
    const int mrow = mt * 16 + ((lane >> 4) << 3);
#pragma unroll
    for (int i = 0; i < 8; ++i) {
        float v = c[i] + bv;
        if (RELU) v = fmaxf(v, 0.0f);
        out[(size_t)(mrow + i) * NOUT + n] = v;
    }
}

// ---------------------------------------------------------------------------
// Launch
// ---------------------------------------------------------------------------
extern "C" void kernel_launch(void* const* d_in, const int* in_sizes, int n_in,
                              void* d_out, int out_size, void* d_ws, size_t ws_size,
                              hipStream_t stream) {
    const float* x   = (const float*)d_in[0];
    const int*   ei  = (const int*)d_in[1];     // [2, N_EDGES]
    const float* ew  = (const float*)d_in[2];
    const float* Wl1 = (const float*)d_in[3];
    const float* bl1 = (const float*)d_in[4];
    const float* Wr1 = (const float*)d_in[5];
    const float* Wl2 = (const float*)d_in[6];
    const float* bl2 = (const float*)d_in[7];
    const float* Wr2 = (const float*)d_in[8];
    float* out = (float*)d_out;

    const int* src = ei;
    const int* dst = ei + N_EDGES;

    // Workspace layout (256B aligned)
    char* ws = (char*)d_ws;
    size_t off = 0;
    auto carve = [&](size_t bytes) {
        char* p = ws + off;
        off += (bytes + 255) & ~(size_t)255;
        return p;
    };
    float* deg  = (float*)carve((size_t)N_NODES * sizeof(float));
    float* agg1 = (float*)carve((size_t)N_NODES * D_IN  * sizeof(float));
    float* h    = (float*)carve((size_t)N_NODES * D_HID * sizeof(float));
    float* agg2 = (float*)carve((size_t)N_NODES * D_HID * sizeof(float));

    hipMemsetAsync(deg,  0, (size_t)N_NODES * sizeof(float), stream);
    hipMemsetAsync(agg1, 0, (size_t)N_NODES * D_IN  * sizeof(float), stream);
    hipMemsetAsync(agg2, 0, (size_t)N_NODES * D_HID * sizeof(float), stream);

    // ---- Layer 1 ----
    {
        long long threads = (long long)N_EDGES * (D_IN >> 2);
        int blocks = (int)((threads + 255) / 256);
        edge_scatter_kernel<<<blocks, 256, 0, stream>>>(x, src, dst, ew,
                                                        agg1, deg, D_IN, 1);
    }
    {
        long long threads = (long long)N_NODES * (D_IN >> 2);
        int blocks = (int)((threads + 255) / 256);
        normalize_kernel<<<blocks, 256, 0, stream>>>(agg1, deg, D_IN);
    }
    {
        int tiles = (N_NODES / 16) * (D_HID / 16);     // 50000
        int blocks = (tiles + 3) / 4;
        sage_gemm_kernel<D_IN, D_HID, true><<<blocks, 128, 0, stream>>>(
            agg1, Wl1, x, Wr1, bl1, h);
    }

    // ---- Layer 2 ----
    {
        long long threads = (long long)N_EDGES * (D_HID >> 2);
        int blocks = (int)((threads + 255) / 256);
        edge_scatter_kernel<<<blocks, 256, 0, stream>>>(h, src, dst, ew,
                                                        agg2, deg, D_HID, 0);
    }
    {
        long long threads = (long long)N_NODES * (D_HID >> 2);
        int blocks = (int)((threads + 255) / 256);
        normalize_kernel<<<blocks, 256, 0, stream>>>(agg2, deg, D_HID);
    }
    {
        int tiles = (N_NODES / 16) * (D_OUT / 16);     // 25000
        int blocks = (tiles + 3) / 4;
        sage_gemm_kernel<D_HID, D_OUT, false><<<blocks, 128, 0, stream>>>(
            agg2, Wl2, h, Wr2, bl2, out);
    }
}